// ROIPooling_5927054868864
// MI455X (gfx1250) — compile-verified
//
#include <hip/hip_runtime.h>
#include <float.h>

#define BATCH   2
#define NROI    32
#define CHAN    256
#define FH      38
#define FW      38
#define OUTP    7
#define CPB     8                    // channels per block
#define CHUNKS  (CHAN / CPB)         // 32 channel chunks
#define NTHR    256                  // 8 wave32s

__global__ __launch_bounds__(NTHR)
void roi_pool_tdm_kernel(const float* __restrict__ feat,
                         const float* __restrict__ rois,
                         float* __restrict__ out) {
    // Worst-case crop is the full 38x38 plane per channel: 8 * 1444 * 4B = 46.2 KB LDS.
    __shared__ float smem[CPB * FH * FW];

    const int blk   = blockIdx.x;
    const int chunk = blk % CHUNKS;
    const int roi   = blk / CHUNKS;          // == b*NROI + r (batch-major, matches reshape)
    const int b     = roi / NROI;

    // Box -> integer crop, exactly like the reference (floor on tl, ceil on br).
    const float* box = rois + (size_t)roi * 4;
    int x1 = (int)floorf(box[0] * (float)FW);
    int y1 = (int)floorf(box[1] * (float)FH);
    int x2 = (int)ceilf (box[2] * (float)FW);
    int y2 = (int)ceilf (box[3] * (float)FH);
    // Defensive clamps (no-ops for valid normalized boxes).
    x1 = max(0, min(x1, FW - 1));
    y1 = max(0, min(y1, FH - 1));
    x2 = max(x1 + 1, min(x2, FW));
    y2 = max(y1 + 1, min(y2, FH));

    const int cw   = x2 - x1;
    const int ch   = y2 - y1;
    const int crop = cw * ch;

    const int c0 = chunk * CPB;
    // Byte address of crop origin (channel c0 of this image).
    const unsigned long long gorg =
        (unsigned long long)(feat + (((size_t)b * CHAN + (size_t)c0) * FH + (size_t)y1) * FW + (size_t)x1);

    // addrspace(3) byte offset of the shared buffer.
    const unsigned lbase = (unsigned)(size_t)&smem[0];

    const int tid = threadIdx.x;

    // ---- Stage the 3-D crop tile (8 x ch x cw) with ONE Tensor Data Mover op ----
    if (tid < 32) {                       // one wave issues the DMA (EXEC ignored by TDM)
        // All descriptor words are block-uniform; force them into SGPRs.
        const int s_lbase = __builtin_amdgcn_readfirstlane((int)lbase);
        const int s_galo  = __builtin_amdgcn_readfirstlane((int)(unsigned)gorg);
        const int s_gahi  = __builtin_amdgcn_readfirstlane(
            (int)(((unsigned)(gorg >> 32) & 0x01FFFFFFu) | 0x80000000u)); // addr[56:32] | type=2
        const int s_dim0  = __builtin_amdgcn_readfirstlane(cw << 16);     // tensor_dim0 lo16 / tile_dim0
        const int s_dim1  = __builtin_amdgcn_readfirstlane(ch << 16);     // tensor_dim1 lo16
        const int s_t12   = __builtin_amdgcn_readfirstlane((CPB << 16) | ch); // tile_dim2 | tile_dim1

        // D# built in pinned, quad-aligned SGPRs:
        //   group0 = s[64:67], group1 = s[68:75], group2 = s[76:79], group3 aliased to group2.
        asm volatile(
            "s_mov_b32 s64, 1\n\t"            // count=1, no gather, user mode
            "s_mov_b32 s65, %0\n\t"           // lds_addr
            "s_mov_b32 s66, %1\n\t"           // global_addr[31:0]
            "s_mov_b32 s67, %2\n\t"           // global_addr[56:32] | type=2
            "s_mov_b32 s68, 0x20000\n\t"      // data_size=2 (4B); mask/pad/iterate=0
            "s_mov_b32 s69, %3\n\t"           // tensor_dim0 lo16 (abar addr = 0)
            "s_mov_b32 s70, %4\n\t"           // tensor_dim0 hi=0 | tensor_dim1 lo16
            "s_mov_b32 s71, %3\n\t"           // tensor_dim1 hi=0 | tile_dim0
            "s_mov_b32 s72, %5\n\t"           // tile_dim1 | tile_dim2
            "s_mov_b32 s73, 38\n\t"           // tensor_dim0_stride = W
            "s_mov_b32 s74, 0x5a40000\n\t"    // stride0 hi=0 | tensor_dim1_stride lo16 = 1444
            "s_mov_b32 s75, 0\n\t"            // tensor_dim1_stride hi
            "s_mov_b32 s76, 8\n\t"            // tensor_dim2 = CPB
            "s_mov_b32 s77, 0\n\t"            // tensor_dim3 (unused)
            "s_mov_b32 s78, 0\n\t"            // tensor_dim2_stride lo (unused)
            "s_mov_b32 s79, 0\n\t"            // stride hi | tile_dim3 = 0
            "tensor_load_to_lds s[64:67], s[68:75], s[76:79], s[76:79]"
            :
            : "s"(s_lbase), "s"(s_galo), "s"(s_gahi), "s"(s_dim0), "s"(s_dim1), "s"(s_t12)
            : "s64","s65","s66","s67","s68","s69","s70","s71",
              "s72","s73","s74","s75","s76","s77","s78","s79",
              "memory");
        __builtin_amdgcn_s_wait_tensorcnt(0);                 // drain TENSORcnt
    }
    __syncthreads();                                          // LDS now holds [CPB][ch][cw]

    // ---- Adaptive 7x7 max-pool over the staged crop ----
    // Output index for this block is contiguous in o: perfect store coalescing.
    for (int o = tid; o < CPB * OUTP * OUTP; o += NTHR) {
        int cl  = o / (OUTP * OUTP);          // constant divisor -> mul/shift
        int bin = o - cl * (OUTP * OUTP);
        int oy  = bin / OUTP;                 // constant divisor
        int ox  = bin - oy * OUTP;

        int ys = (oy * ch) / OUTP;
        int ye = ((oy + 1) * ch + OUTP - 1) / OUTP;   // ceil
        int xs = (ox * cw) / OUTP;
        int xe = ((ox + 1) * cw + OUTP - 1) / OUTP;   // ceil

        const float* sp = smem + (size_t)cl * crop;
        float m = -FLT_MAX;
        for (int y = ys; y < ye; ++y) {
            const float* row = sp + (size_t)y * cw;
            for (int x = xs; x < xe; ++x)
                m = fmaxf(m, row[x]);
        }

        out[((size_t)roi * CHAN + (size_t)c0) * (OUTP * OUTP) + (size_t)o] = m;
    }
}

extern "C" void kernel_launch(void* const* d_in, const int* in_sizes, int n_in,
                              void* d_out, int out_size, void* d_ws, size_t ws_size,
                              hipStream_t stream) {
    const float* feat = (const float*)d_in[0];   // [2,256,38,38] f32
    const float* rois = (const float*)d_in[1];   // [2,32,4] f32
    float* out = (float*)d_out;                  // [64,256,7,7] f32

    dim3 grid(BATCH * NROI * CHUNKS);            // 2048 blocks
    roi_pool_tdm_kernel<<<grid, NTHR, 0, stream>>>(feat, rois, out);
}